// GCN_11647951307429
// MI455X (gfx1250) — compile-verified
//
#include <hip/hip_runtime.h>
#include <hip/hip_bf16.h>
#include <math.h>

typedef float v2f __attribute__((ext_vector_type(2)));
typedef float v8f __attribute__((ext_vector_type(8)));

#define DHID 128
#define NGRAPHS 64

// ---------------- degree / dinv ----------------
__global__ void deg_init_kernel(float* __restrict__ deg, int N) {
    int i = blockIdx.x * blockDim.x + threadIdx.x;
    if (i < N) deg[i] = 1.0f;  // self loop
}

__global__ void deg_count_kernel(float* __restrict__ deg, const int* __restrict__ col, int E) {
    int e = blockIdx.x * blockDim.x + threadIdx.x;
    if (e < E) atomicAdd(&deg[col[e]], 1.0f);
}

__global__ void deg_finalize_kernel(float* __restrict__ deg, int N) {
    int i = blockIdx.x * blockDim.x + threadIdx.x;
    if (i < N) deg[i] = rsqrtf(deg[i]);  // deg >= 1 always
}

// ---------------- dense GEMM: H[N x 128] = X[N x 128] @ W[128 x 128] ------
// fp32 WMMA 16x16x4. One block = 16 rows of output; 8 waves, one 16-col tile
// per wave; K-loop of 32 chained WMMAs. N must be a multiple of 16.
__global__ __launch_bounds__(256) void gemm_n128_kernel(const float* __restrict__ X,
                                                        const float* __restrict__ W,
                                                        float* __restrict__ H) {
    __shared__ float sA[16 * DHID];  // 8 KB A tile, shared by all 8 waves
    const int tid  = threadIdx.x;
    const int m0   = blockIdx.x * 16;

    // cooperative load of the 16x128 A tile (2048 floats, 512 float4)
    {
        const float4* src = (const float4*)(X + (size_t)m0 * DHID);
        float4* dst = (float4*)sA;
        for (int i = tid; i < 16 * DHID / 4; i += 256) dst[i] = src[i];
    }
    __syncthreads();

    const int wave = tid >> 5;
    const int lane = tid & 31;
    const int half = lane >> 4;   // 0: K=k0..k0+1  1: K=k0+2..k0+3
    const int ml   = lane & 15;   // row (A) / col (B) index within tile
    const int n0   = wave * 16;

    v8f c = {};
    #pragma unroll
    for (int k0 = 0; k0 < DHID; k0 += 4) {
        const int kk = k0 + half * 2;
        v2f a, b;
        a.x = sA[ml * DHID + kk];
        a.y = sA[ml * DHID + kk + 1];
        b.x = W[kk * DHID + n0 + ml];
        b.y = W[(kk + 1) * DHID + n0 + ml];
        c = __builtin_amdgcn_wmma_f32_16x16x4_f32(false, a, false, b,
                                                  (short)0, c, false, false);
    }

    // C/D layout: VGPR r -> M = r (lanes 0-15) / r+8 (lanes 16-31), N = ml
    #pragma unroll
    for (int r = 0; r < 8; ++r) {
        const int m = m0 + half * 8 + r;
        H[(size_t)m * DHID + n0 + ml] = c[r];
    }
}

// ---------------- self-loop init: OUT[n,f] = H[n,f] * dinv[n]^2 -----------
__global__ void selfloop_init_kernel(const float* __restrict__ H,
                                     const float* __restrict__ dinv,
                                     float* __restrict__ OUT, int total) {
    int idx = blockIdx.x * blockDim.x + threadIdx.x;
    if (idx >= total) return;
    float di = dinv[idx >> 7];
    OUT[idx] = H[idx] * di * di;
}

// ---------------- edge scatter: one wave per edge -------------------------
__global__ __launch_bounds__(256) void scatter_edges_kernel(const float* __restrict__ H,
                                                            float* __restrict__ OUT,
                                                            const int* __restrict__ row,
                                                            const int* __restrict__ col,
                                                            const float* __restrict__ dinv,
                                                            int E) {
    const int wid  = (int)((blockIdx.x * (unsigned)blockDim.x + threadIdx.x) >> 5);
    const int lane = threadIdx.x & 31;
    if (wid >= E) return;
    const int s = row[wid];
    const int d = col[wid];
    const float w = dinv[s] * dinv[d];
    const float4 v = ((const float4*)(H + (size_t)s * DHID))[lane];  // 512B/wave
    float* dst = OUT + (size_t)d * DHID + lane * 4;
    atomicAdd(dst + 0, v.x * w);
    atomicAdd(dst + 1, v.y * w);
    atomicAdd(dst + 2, v.z * w);
    atomicAdd(dst + 3, v.w * w);
}

// ---------------- bias (+ optional ELU) -----------------------------------
__global__ void bias_act_kernel(float* __restrict__ H, const float* __restrict__ bias,
                                int total, int doElu) {
    int idx = blockIdx.x * blockDim.x + threadIdx.x;
    if (idx >= total) return;
    float v = H[idx] + bias[idx & (DHID - 1)];
    if (doElu) v = v > 0.0f ? v : (expf(v) - 1.0f);
    H[idx] = v;
}

// ---------------- zero scratch --------------------------------------------
__global__ void zero_kernel(float* __restrict__ p, int n) {
    int i = blockIdx.x * blockDim.x + threadIdx.x;
    if (i < n) p[i] = 0.0f;
}

// ---------------- pooled accumulation (adds b2 on the way) ----------------
__global__ void pool_accum_kernel(const float* __restrict__ H,
                                  const float* __restrict__ b2,
                                  const int* __restrict__ batch,
                                  float* __restrict__ psum,
                                  float* __restrict__ pcnt, int total) {
    int idx = blockIdx.x * blockDim.x + threadIdx.x;
    if (idx >= total) return;
    const int n = idx >> 7, f = idx & (DHID - 1);
    const int g = batch[n];
    atomicAdd(&psum[g * DHID + f], H[idx] + b2[f]);
    if (f == 0) atomicAdd(&pcnt[g], 1.0f);
}

// ---------------- head: mean -> fc1+relu -> fc2 -> log_softmax ------------
__global__ __launch_bounds__(256) void head_kernel(const float* __restrict__ psum,
                                                   const float* __restrict__ pcnt,
                                                   const float* __restrict__ fc1W,
                                                   const float* __restrict__ fc1b,
                                                   const float* __restrict__ fc2W,
                                                   const float* __restrict__ fc2b,
                                                   float* __restrict__ out) {
    __shared__ float g[NGRAPHS * DHID];   // 32 KB
    __shared__ float mid[NGRAPHS * 20];
    __shared__ float lo[NGRAPHS * 10];
    const int tid = threadIdx.x;

    for (int i = tid; i < NGRAPHS * DHID; i += 256) {
        float c = pcnt[i >> 7];
        g[i] = psum[i] / fmaxf(c, 1.0f);
    }
    __syncthreads();

    for (int i = tid; i < NGRAPHS * 20; i += 256) {
        const int gr = i / 20, j = i % 20;
        float s = fc1b[j];
        const float* gg = &g[gr * DHID];
        for (int k = 0; k < DHID; ++k) s += gg[k] * fc1W[k * 20 + j];
        mid[i] = fmaxf(s, 0.0f);
    }
    __syncthreads();

    for (int i = tid; i < NGRAPHS * 10; i += 256) {
        const int gr = i / 10, j = i % 10;
        float s = fc2b[j];
        for (int k = 0; k < 20; ++k) s += mid[gr * 20 + k] * fc2W[k * 10 + j];
        lo[i] = s;
    }
    __syncthreads();

    if (tid < NGRAPHS) {
        float mx = -INFINITY;
        for (int j = 0; j < 10; ++j) mx = fmaxf(mx, lo[tid * 10 + j]);
        float se = 0.0f;
        for (int j = 0; j < 10; ++j) se += expf(lo[tid * 10 + j] - mx);
        const float lse = mx + logf(se);
        for (int j = 0; j < 10; ++j) out[tid * 10 + j] = lo[tid * 10 + j] - lse;
    }
}

// ---------------- launch --------------------------------------------------
extern "C" void kernel_launch(void* const* d_in, const int* in_sizes, int n_in,
                              void* d_out, int out_size, void* d_ws, size_t ws_size,
                              hipStream_t stream) {
    const float* x     = (const float*)d_in[0];
    const int*   ei    = (const int*)d_in[1];   // [2, E]: row = ei, col = ei + E
    const int*   batch = (const int*)d_in[2];
    const float* W1    = (const float*)d_in[3];
    const float* b1    = (const float*)d_in[4];
    const float* W2    = (const float*)d_in[5];
    const float* b2    = (const float*)d_in[6];
    const float* fc1W  = (const float*)d_in[7];
    const float* fc1b  = (const float*)d_in[8];
    const float* fc2W  = (const float*)d_in[9];
    const float* fc2b  = (const float*)d_in[10];

    const int N = in_sizes[0] / DHID;   // 100000
    const int E = in_sizes[1] / 2;      // 1600000
    const int total = N * DHID;

    float* bufA = (float*)d_ws;                    // N*128
    float* bufB = bufA + (size_t)N * DHID;         // N*128
    float* dinv = bufB + (size_t)N * DHID;         // N
    float* psum = dinv + N;                        // 64*128
    float* pcnt = psum + NGRAPHS * DHID;           // 64

    const int T = 256;
    const int gN   = (N + T - 1) / T;
    const int gE   = (E + T - 1) / T;
    const int gTot = (total + T - 1) / T;
    const int gEW  = (int)(((long long)E * 32 + T - 1) / T);

    // degree -> dinv
    deg_init_kernel<<<gN, T, 0, stream>>>(dinv, N);
    deg_count_kernel<<<gE, T, 0, stream>>>(dinv, ei + E, E);
    deg_finalize_kernel<<<gN, T, 0, stream>>>(dinv, N);

    // layer 1: h = x @ W1 ; aggregate ; +b1 ; ELU
    gemm_n128_kernel<<<N / 16, T, 0, stream>>>(x, W1, bufA);
    selfloop_init_kernel<<<gTot, T, 0, stream>>>(bufA, dinv, bufB, total);
    scatter_edges_kernel<<<gEW, T, 0, stream>>>(bufA, bufB, ei, ei + E, dinv, E);
    bias_act_kernel<<<gTot, T, 0, stream>>>(bufB, b1, total, 1);

    // layer 2: h = h1 @ W2 ; aggregate (b2 folded into pooling)
    gemm_n128_kernel<<<N / 16, T, 0, stream>>>(bufB, W2, bufA);
    selfloop_init_kernel<<<gTot, T, 0, stream>>>(bufA, dinv, bufB, total);
    scatter_edges_kernel<<<gEW, T, 0, stream>>>(bufA, bufB, ei, ei + E, dinv, E);

    // pooling + head
    zero_kernel<<<(NGRAPHS * DHID + NGRAPHS + T - 1) / T, T, 0, stream>>>(psum, NGRAPHS * DHID + NGRAPHS);
    pool_accum_kernel<<<gTot, T, 0, stream>>>(bufB, b2, batch, psum, pcnt, total);
    head_kernel<<<1, T, 0, stream>>>(psum, pcnt, fc1W, fc1b, fc2W, fc2b, (float*)d_out);
}